// WordEncoder_55997783605744
// MI455X (gfx1250) — compile-verified
//
#include <hip/hip_runtime.h>
#include <math.h>

typedef __attribute__((ext_vector_type(2))) float v2f;
typedef __attribute__((ext_vector_type(8))) float v8f;

#define DIMS 768   // feature dim, 192 float4 per row, 6 float4 per lane (wave32)

// ---------------------------------------------------------------------------
// Phase 1: scores[i] = dot(x[i,:], w) + bias   via V_WMMA_F32_16X16X4_F32
// One wave handles a 16-row tile. A-frag (16x4 f32, ISA layout): lane = half*16+m
// holds x[rowBase+m][k0+2*half] in v0 and x[rowBase+m][k0+2*half+1] in v1, i.e.
// a single float2 load. B-frag replicates w[k] across all 16 columns, which by
// the symmetric 4x16 layout is the *same* float2 pattern on attn_w. After the
// K-loop, D column 0 (lane 0 rows 0..7, lane 16 rows 8..15, VGPRs 0..7) holds
// the 16 dot products.
// ---------------------------------------------------------------------------
__global__ __launch_bounds__(256) void we_scores_wmma(
    const float* __restrict__ x, const float* __restrict__ w,
    const float* __restrict__ bias, float* __restrict__ scores, int n_sub)
{
    const int lane  = threadIdx.x & 31;
    const int wave  = threadIdx.x >> 5;
    const int tile  = blockIdx.x * (blockDim.x >> 5) + wave;
    const int rowBase = tile * 16;
    if (rowBase >= n_sub) return;                  // wave-uniform: EXEC stays all-1s

    const int half = lane >> 4;                    // 0: K={0,1}, 1: K={2,3}
    const int m    = lane & 15;                    // row within tile
    const float* rp = x + (size_t)(rowBase + m) * DIMS + 2 * half;
    const float* wp = w + 2 * half;

    v8f c = {};
#if __has_builtin(__builtin_amdgcn_wmma_f32_16x16x4_f32)
    #pragma unroll 8
    for (int k0 = 0; k0 < DIMS; k0 += 4) {
        v2f a  = *(const v2f*)(rp + k0);           // A[m][k0+2h], A[m][k0+2h+1]
        v2f bb = *(const v2f*)(wp + k0);           // w replicated across N columns
        c = __builtin_amdgcn_wmma_f32_16x16x4_f32(
                /*neg_a=*/false, a, /*neg_b=*/false, bb,
                /*c_mod=*/(short)0, c, /*reuse_a=*/false, /*reuse_b=*/false);
    }
    const float b0 = bias[0];
    if (m == 0) {                                  // lanes 0 and 16 hold column N=0
        float* sp = scores + rowBase + 8 * half;
        sp[0] = c[0] + b0; sp[1] = c[1] + b0; sp[2] = c[2] + b0; sp[3] = c[3] + b0;
        sp[4] = c[4] + b0; sp[5] = c[5] + b0; sp[6] = c[6] + b0; sp[7] = c[7] + b0;
    }
#else
    // Fallback: per-lane partial dot over this half's K residues, combine halves.
    float acc = 0.f;
    #pragma unroll 8
    for (int k0 = 0; k0 < DIMS; k0 += 4)
        acc = fmaf(rp[k0], wp[k0], fmaf(rp[k0 + 1], wp[k0 + 1], acc));
    acc += __shfl_xor(acc, 16, 32);
    if (half == 0) scores[rowBase + m] = acc + bias[0];
#endif
}

// ---------------------------------------------------------------------------
// Phase 2: one wave per word. Softmax over scores[start:end) (spans are tiny),
// then out[w,:] = sum_j p_j * x[j,:].  6 float4 accumulators/lane cover 768
// dims with b128 loads/stores; prefetch next row while accumulating.
// ---------------------------------------------------------------------------
__global__ __launch_bounds__(256) void we_softmax_gather(
    const float* __restrict__ x, const float* __restrict__ scores,
    const int* __restrict__ wb, float* __restrict__ out, int n_words)
{
    const int lane = threadIdx.x & 31;
    const int wid  = (int)((blockIdx.x * blockDim.x + threadIdx.x) >> 5);
    if (wid >= n_words) return;

    const int start = wb[2 * wid];
    const int end   = wb[2 * wid + 1];

    float mx = -INFINITY;
    for (int j = start; j < end; ++j) mx = fmaxf(mx, scores[j]);
    float denom = 0.f;
    for (int j = start; j < end; ++j) denom += expf(scores[j] - mx);
    const float inv = 1.0f / denom;

    float4 a0 = {0,0,0,0}, a1 = {0,0,0,0}, a2 = {0,0,0,0};
    float4 a3 = {0,0,0,0}, a4 = {0,0,0,0}, a5 = {0,0,0,0};

    for (int j = start; j < end; ++j) {
        if (j + 1 < end)
            __builtin_prefetch(x + (size_t)(j + 1) * DIMS + lane * 4, 0, 0);
        const float p = expf(scores[j] - mx) * inv;
        const float4* xp = (const float4*)(x + (size_t)j * DIMS);
        float4 v;
        v = xp[0*32 + lane]; a0.x += p*v.x; a0.y += p*v.y; a0.z += p*v.z; a0.w += p*v.w;
        v = xp[1*32 + lane]; a1.x += p*v.x; a1.y += p*v.y; a1.z += p*v.z; a1.w += p*v.w;
        v = xp[2*32 + lane]; a2.x += p*v.x; a2.y += p*v.y; a2.z += p*v.z; a2.w += p*v.w;
        v = xp[3*32 + lane]; a3.x += p*v.x; a3.y += p*v.y; a3.z += p*v.z; a3.w += p*v.w;
        v = xp[4*32 + lane]; a4.x += p*v.x; a4.y += p*v.y; a4.z += p*v.z; a4.w += p*v.w;
        v = xp[5*32 + lane]; a5.x += p*v.x; a5.y += p*v.y; a5.z += p*v.z; a5.w += p*v.w;
    }

    float4* op = (float4*)(out + (size_t)wid * DIMS);
    op[0*32 + lane] = a0; op[1*32 + lane] = a1; op[2*32 + lane] = a2;
    op[3*32 + lane] = a3; op[4*32 + lane] = a4; op[5*32 + lane] = a5;
}

// ---------------------------------------------------------------------------
// Inputs (setup_inputs order): x f32[16384*768], attn_w f32[768], attn_b f32[1],
// word_boundaries i32[8192*2]. Output: f32[8192*768]. Scratch: scores (64 KB).
// ---------------------------------------------------------------------------
extern "C" void kernel_launch(void* const* d_in, const int* in_sizes, int n_in,
                              void* d_out, int out_size, void* d_ws, size_t ws_size,
                              hipStream_t stream) {
    const float* x  = (const float*)d_in[0];
    const float* w  = (const float*)d_in[1];
    const float* b  = (const float*)d_in[2];
    const int*   wb = (const int*)d_in[3];
    float* out    = (float*)d_out;
    float* scores = (float*)d_ws;

    const int n_sub   = in_sizes[0] / DIMS;
    const int n_words = in_sizes[3] / 2;

    const int tiles   = (n_sub + 15) / 16;     // one wave per 16-row tile
    const int wpb     = 8;                     // 8 waves (256 threads) per block
    const int blocks1 = (tiles + wpb - 1) / wpb;
    we_scores_wmma<<<blocks1, wpb * 32, 0, stream>>>(x, w, b, scores, n_sub);

    const int blocks2 = (n_words + wpb - 1) / wpb;  // one wave per word
    we_softmax_gather<<<blocks2, wpb * 32, 0, stream>>>(x, scores, wb, out, n_words);
}